// FFTConv_85779086835962
// MI455X (gfx1250) — compile-verified
//
#include <hip/hip_runtime.h>
#include <math.h>

// ---------------------------------------------------------------------------
// FFT long-convolution (Hyena/S4 style) + skip + exact GELU
//   x:(B=8,H=1024,L=8192)  k:(C=1,H,L)  D:(C,H)  ->  out:(B,H,L) fp32
// N = 2L = 16384 zero-padded FFT length.
//
// One 1024-thread workgroup (32 wave32) per (b,h) row, 2 WGs per WGP:
//   - CDNA5 async global->LDS copies (global_load_async_to_lds_b128) of the
//     x-row and k-row, twiddle table built while copies are in flight
//   - pack z = x + i*k  ->  ONE radix-4 DIF forward FFT (7 stages, in place,
//     digit-reversed output, no bit-reversal pass)
//   - product spectrum directly in scrambled order via the real-packing
//     identity  X*K = (Z[f]^2 - conj(Z[N-f])^2)/(4i); the f <-> N-f partner
//     position is rev4((N - rev4(p)) mod N)  (rev4 is an involution)
//   - radix-4 DIT inverse FFT back to natural order
//   - y = Re(ifft)[0:L] + D[h]*x -> exact-erf GELU -> nontemporal store
// LDS: Z (128K) + quarter-wave twiddle table (32K) = 160 KB  (2 blocks/WGP).
// ---------------------------------------------------------------------------

#define FFT_N     16384
#define FFT_NH    8192          // N/2
#define FFT_NQ    4096          // N/4 quarter-wave table
#define QSHIFT    12            // log2(FFT_NQ)
#define SEQ_L     8192
#define NTHREADS  1024
#define NHEADS    1024

#if defined(__AMDGCN__) && \
    __has_builtin(__builtin_amdgcn_global_load_async_to_lds_b128) && \
    __has_builtin(__builtin_amdgcn_s_wait_asynccnt)
#define USE_ASYNC_LDS 1
#else
#define USE_ASYNC_LDS 0
#endif

#if USE_ASYNC_LDS
typedef int v4i_cdna5 __attribute__((ext_vector_type(4)));
typedef __attribute__((address_space(1))) v4i_cdna5 as1_v4i;   // global
typedef __attribute__((address_space(3))) v4i_cdna5 as3_v4i;   // LDS
#endif

__device__ __forceinline__ float2 cmulf(float2 a, float2 b) {
  return make_float2(a.x * b.x - a.y * b.y, a.x * b.y + a.y * b.x);
}

// base-4 digit reversal of a 14-bit index (7 digits): bit-reverse, then swap
// adjacent bit pairs.  Involution.
__device__ __forceinline__ int rev4_14(int p) {
  unsigned r = __brev((unsigned)p) >> 18;
  return (int)(((r & 0x1555u) << 1) | ((r & 0x2AAAu) >> 1));
}

// exp(-2*pi*i*j/FFT_N) for 0 <= j < FFT_N, from quarter table
// tab[j] = (cos(2*pi*j/N), sin(2*pi*j/N)), 0 <= j < N/4.
__device__ __forceinline__ float2 twd(const float2* __restrict__ tab, int j) {
  float2 t = tab[j & (FFT_NQ - 1)];
  int q = (j >> QSHIFT) & 3;
  float2 r;
  if (q == 0)      r = make_float2( t.x, -t.y);
  else if (q == 1) r = make_float2(-t.y, -t.x);
  else if (q == 2) r = make_float2(-t.x,  t.y);
  else             r = make_float2( t.y,  t.x);
  return r;
}

__device__ __forceinline__ void g2l_b128(const float* __restrict__ g, float* l) {
#if USE_ASYNC_LDS
  __builtin_amdgcn_global_load_async_to_lds_b128(
      (as1_v4i*)g, (as3_v4i*)l, 0, 0);
#else
  *(float4*)l = *(const float4*)g;
#endif
}

__global__ __launch_bounds__(NTHREADS)
void fftconv_kernel(const float* __restrict__ x, const float* __restrict__ kern,
                    const float* __restrict__ Dw, float* __restrict__ out)
{
  extern __shared__ float2 smem[];
  float2* Xs  = smem;                  // [FFT_N]  z spectrum / ifft buffer
  float2* tab = smem + FFT_N;          // [FFT_NQ] quarter-wave twiddles
  float*  XsF = (float*)Xs;

  const int tid = threadIdx.x;
  const int bid = blockIdx.x;          // bid = h*8 + b : 8 consecutive blocks share a k row (L2 reuse)
  const int h = bid >> 3;
  const int b = bid & 7;
  const long rowX = ((long)b * NHEADS + h) * (long)SEQ_L;
  const long rowK = (long)h * (long)SEQ_L;

  // ---- async-stage both rows as packed floats into Xs' upper half ----
  // floats [FFT_N, FFT_N+L)      : x row
  // floats [FFT_N+L, FFT_N+2L)   : k row
#pragma unroll
  for (int i = 0; i < 2; ++i) {
    const int fo = (tid + i * NTHREADS) * 4;            // 16B-aligned float offset
    g2l_b128(x    + rowX + fo, XsF + FFT_N + fo);
    g2l_b128(kern + rowK + fo, XsF + FFT_N + SEQ_L + fo);
  }

  // ---- build twiddle table while the copies are in flight ----
#pragma unroll
  for (int i = 0; i < FFT_NQ / NTHREADS; ++i) {
    const int j = tid + i * NTHREADS;
    float s, c;
    sincosf(6.283185307179586476f * (float)j / (float)FFT_N, &s, &c);
    tab[j] = make_float2(c, s);
  }

#if USE_ASYNC_LDS
  __builtin_amdgcn_s_wait_asynccnt(0);
#endif
  __syncthreads();

  // keep this thread's x slice in registers for the skip connection
  float xreg[8], kreg[8];
#pragma unroll
  for (int s = 0; s < 8; ++s) {
    const int l = tid + s * NTHREADS;
    xreg[s] = XsF[FFT_N + l];
    kreg[s] = XsF[FFT_N + SEQ_L + l];
  }
  __syncthreads();

  // pack z = x + i*k, zero-pad upper half (overwrites the staging region)
#pragma unroll
  for (int s = 0; s < 8; ++s) {
    const int l = tid + s * NTHREADS;
    Xs[l]          = make_float2(xreg[s], kreg[s]);
    Xs[FFT_NH + l] = make_float2(0.f, 0.f);
  }
  __syncthreads();

  // ---- forward radix-4 DIF: 7 stages, digit-reversed output ----
  for (int lm = 12; lm >= 0; lm -= 2) {
    const int m = 1 << lm;                  // quarter-span
    const int F = 1 << (QSHIFT - lm);       // N / (4m)
#pragma unroll
    for (int it = 0; it < 4; ++it) {
      const int j   = tid + it * NTHREADS;  // butterfly id < N/4
      const int off = j & (m - 1);
      const int i0  = 4 * j - 3 * off;
      const int i1 = i0 + m, i2 = i1 + m, i3 = i2 + m;
      const float2 W1 = twd(tab, off * F);
      const float2 W2 = cmulf(W1, W1);
      const float2 W3 = cmulf(W2, W1);
      float2 z0 = Xs[i0], z1 = Xs[i1], z2 = Xs[i2], z3 = Xs[i3];
      float2 t0 = make_float2(z0.x + z2.x, z0.y + z2.y);
      float2 t1 = make_float2(z0.x - z2.x, z0.y - z2.y);
      float2 t2 = make_float2(z1.x + z3.x, z1.y + z3.y);
      float2 t3 = make_float2(z1.x - z3.x, z1.y - z3.y);
      Xs[i0] = make_float2(t0.x + t2.x, t0.y + t2.y);
      Xs[i1] = cmulf(make_float2(t1.x + t3.y, t1.y - t3.x), W1); // (t1 - i t3) W1
      Xs[i2] = cmulf(make_float2(t0.x - t2.x, t0.y - t2.y), W2);
      Xs[i3] = cmulf(make_float2(t1.x - t3.y, t1.y + t3.x), W3); // (t1 + i t3) W3
    }
    __syncthreads();
  }

  // ---- product spectrum in scrambled order via real-packing identity ----
  // position p holds Z[rev4(p)]; partner Z[N-f] sits at rev4((N-f) mod N).
  // Y = X*K = (Z^2 - conj(Zp)^2)/(4i); fold 1/(4N) and the /i rotation in.
  const float scale = 0.25f / (float)FFT_N;
  float2 Yreg[16];
#pragma unroll
  for (int it = 0; it < 16; ++it) {
    const int p  = tid + it * NTHREADS;
    const int f  = rev4_14(p);
    const int pp = rev4_14((FFT_N - f) & (FFT_N - 1));
    const float2 zp = Xs[p];                         // Z[f]
    const float2 zq = Xs[pp];                        // Z[N-f]
    const float2 C  = make_float2(zq.x, -zq.y);      // conj(Z[N-f])
    const float2 z2 = make_float2(zp.x * zp.x - zp.y * zp.y, 2.f * zp.x * zp.y);
    const float2 c2 = make_float2(C.x * C.x - C.y * C.y,     2.f * C.x * C.y);
    const float2 A  = make_float2(z2.x - c2.x, z2.y - c2.y); // Z^2 - C^2
    Yreg[it] = make_float2(A.y * scale, -A.x * scale);       // A/(4i) * 1/N
  }
  __syncthreads();
#pragma unroll
  for (int it = 0; it < 16; ++it) {
    Xs[tid + it * NTHREADS] = Yreg[it];
  }
  __syncthreads();

  // ---- inverse radix-4 DIT: digit-reversed in -> natural out ----
  for (int lm = 0; lm <= 12; lm += 2) {
    const int m = 1 << lm;
    const int F = 1 << (QSHIFT - lm);
#pragma unroll
    for (int it = 0; it < 4; ++it) {
      const int j   = tid + it * NTHREADS;
      const int off = j & (m - 1);
      const int i0  = 4 * j - 3 * off;
      const int i1 = i0 + m, i2 = i1 + m, i3 = i2 + m;
      float2 W1 = twd(tab, off * F);
      W1.y = -W1.y;                          // conjugate for inverse
      const float2 W2 = cmulf(W1, W1);
      const float2 W3 = cmulf(W2, W1);
      float2 z0 = Xs[i0];
      float2 z1 = cmulf(Xs[i1], W1);
      float2 z2 = cmulf(Xs[i2], W2);
      float2 z3 = cmulf(Xs[i3], W3);
      float2 t0 = make_float2(z0.x + z2.x, z0.y + z2.y);
      float2 t1 = make_float2(z0.x - z2.x, z0.y - z2.y);
      float2 t2 = make_float2(z1.x + z3.x, z1.y + z3.y);
      float2 t3 = make_float2(z1.x - z3.x, z1.y - z3.y);
      Xs[i0] = make_float2(t0.x + t2.x, t0.y + t2.y);
      Xs[i1] = make_float2(t1.x - t3.y, t1.y + t3.x);  // t1 + i t3
      Xs[i2] = make_float2(t0.x - t2.x, t0.y - t2.y);
      Xs[i3] = make_float2(t1.x + t3.y, t1.y - t3.x);  // t1 - i t3
    }
    __syncthreads();
  }

  // ---- skip connection + exact GELU + streaming store ----
  const float Dh = Dw[h];
#pragma unroll
  for (int s = 0; s < 8; ++s) {
    const int l = tid + s * NTHREADS;
    const float v = Xs[l].x + Dh * xreg[s];
    const float g = 0.5f * v * (1.0f + erff(v * 0.70710678118654752440f));
    __builtin_nontemporal_store(g, out + rowX + l);
  }
}

extern "C" void kernel_launch(void* const* d_in, const int* in_sizes, int n_in,
                              void* d_out, int out_size, void* d_ws, size_t ws_size,
                              hipStream_t stream) {
  (void)in_sizes; (void)n_in; (void)out_size; (void)d_ws; (void)ws_size;
  const float* x  = (const float*)d_in[0];   // (8, 1024, 8192)
  const float* k  = (const float*)d_in[1];   // (1, 1024, 8192)
  const float* Dw = (const float*)d_in[2];   // (1, 1024)
  float* out = (float*)d_out;                // (8, 1024, 8192)

  const dim3 grid(8 * NHEADS);               // one block per (b,h) row, bid = h*8 + b
  const dim3 block(NTHREADS);
  const size_t shmem = (size_t)(FFT_N + FFT_NQ) * sizeof(float2); // 160 KB
  fftconv_kernel<<<grid, block, shmem, stream>>>(x, k, Dw, out);
}